// PositionalEncoding_61151744360729
// MI455X (gfx1250) — compile-verified
//
#include <hip/hip_runtime.h>
#include <stdint.h>

// Problem constants from the reference: x[B, S=1024, N=25, 3] fp32.
#define NODES   25
#define SEQ     1024
#define FPT     64                                  // frames per tile (per block)
#define TILES_PER_ROW (SEQ / FPT)                   // 16 (power of two!)
#define TRIPLES_PER_TILE (FPT * NODES)              // 1600 node-frames per tile
#define INNER_TRIPLES (TRIPLES_PER_TILE - NODES)    // 1575: triples with in-tile next
#define LOAD_TRIPLES  (TRIPLES_PER_TILE + NODES)    // +1 halo frame = 1625
#define BLOCK   256                                 // 8 waves (wave32)

typedef __attribute__((address_space(3))) float lds_f32;

__global__ __launch_bounds__(BLOCK)
void skel_dist_kernel(const float* __restrict__ x, float* __restrict__ out)
{
    // 1625 triples * 3 floats = 4875 floats = 19.5 KB of LDS
    __shared__ float sh[LOAD_TRIPLES * 3];

    const int bid  = blockIdx.x;
    const int tile = bid & (TILES_PER_ROW - 1);          // s_and, no division
    const bool hasHalo = tile != (TILES_PER_ROW - 1);

    const int tid = threadIdx.x;

    // Each tile covers FPT*NODES*3 = 4800 floats = 19200 bytes (16B-aligned).
    const uint64_t baseElem = (uint64_t)bid * (uint64_t)(FPT * NODES * 3);
    const uint64_t gbase = (uint64_t)(uintptr_t)x + baseElem * 4ull;

    const int nLoadF = hasHalo ? (LOAD_TRIPLES * 3) : (TRIPLES_PER_TILE * 3);
    const int nVec4  = nLoadF >> 2;      // 1218 (halo) or 1200 (no halo)
    const int nTail  = nLoadF & 3;       // 3 (halo) or 0

    lds_f32* sh3 = (lds_f32*)sh;

    // ---- Stage tile (+halo) into LDS with CDNA5 async global->LDS DMA ----
    // Bulk: 128-bit async loads (base and LDS destination both 16B-aligned).
    for (int i4 = tid; i4 < nVec4; i4 += BLOCK) {
        uint32_t loff = (uint32_t)(uintptr_t)(sh3 + i4 * 4);
        uint32_t goff = (uint32_t)i4 * 16u;
        asm volatile("global_load_async_to_lds_b128 %0, %1, %2"
                     :
                     : "v"(loff), "v"(goff), "s"(gbase)
                     : "memory");
    }
    // Tail (3 floats, halo tiles only): 32-bit async loads.
    if (tid < nTail) {
        const int i = (nVec4 << 2) + tid;
        uint32_t loff = (uint32_t)(uintptr_t)(sh3 + i);
        uint32_t goff = (uint32_t)i * 4u;
        asm volatile("global_load_async_to_lds_b32 %0, %1, %2"
                     :
                     : "v"(loff), "v"(goff), "s"(gbase)
                     : "memory");
    }
#if __has_builtin(__builtin_amdgcn_s_wait_asynccnt)
    __builtin_amdgcn_s_wait_asynccnt(0);
#else
    asm volatile("s_wait_asynccnt 0x0" ::: "memory");
#endif
    __syncthreads();   // other waves' async loads also feed our LDS reads

    // ---- Compute: d = ||frame[s+1] - frame[s]||_2 per node; out = x + d ----
    // Triples below `limit` have a valid "next" frame staged in LDS;
    // the rest (final frame of the sequence) use d = 0.
    const int limit = hasHalo ? TRIPLES_PER_TILE : INNER_TRIPLES;

    float* o = out + baseElem;
    for (int t = tid; t < TRIPLES_PER_TILE; t += BLOCK) {
        const float cx = sh[3*t + 0];
        const float cy = sh[3*t + 1];
        const float cz = sh[3*t + 2];
        float nx = cx, ny = cy, nz = cz;          // default: last frame, d = 0
        if (t < limit) {
            const int u = t + NODES;              // same node, next frame
            nx = sh[3*u + 0];
            ny = sh[3*u + 1];
            nz = sh[3*u + 2];
        }
        const float dx = nx - cx, dy = ny - cy, dz = nz - cz;
        const float d  = sqrtf(dx*dx + dy*dy + dz*dz);
        // Output is streamed, never re-read: non-temporal stores keep L2 clean.
        __builtin_nontemporal_store(cx + d, o + 3*t + 0);
        __builtin_nontemporal_store(cy + d, o + 3*t + 1);
        __builtin_nontemporal_store(cz + d, o + 3*t + 2);
    }
}

extern "C" void kernel_launch(void* const* d_in, const int* in_sizes, int n_in,
                              void* d_out, int out_size, void* d_ws, size_t ws_size,
                              hipStream_t stream) {
    const float* x = (const float*)d_in[0];
    float* out     = (float*)d_out;

    const int total  = in_sizes[0];                 // B * S * N * 3
    const int rows   = total / (SEQ * NODES * 3);   // B (512 for reference)
    const int blocks = rows * TILES_PER_ROW;        // 8192 blocks

    hipLaunchKernelGGL(skel_dist_kernel, dim3(blocks), dim3(BLOCK), 0, stream,
                       x, out);
}